// NTMCell_84232898609680
// MI455X (gfx1250) — compile-verified
//
#include <hip/hip_runtime.h>
#include <math.h>

// Problem sizes (fixed by the reference)
#define NN 4096
#define SS 128
#define FF 128
#define DD 256

typedef __attribute__((ext_vector_type(2))) float v2f;
typedef __attribute__((ext_vector_type(8))) float v8f;

__device__ __forceinline__ v8f wmma4(v2f a, v2f b, v8f c) {
    // V_WMMA_F32_16X16X4_F32 : D = A(16x4) * B(4x16) + C(16x16), fp32
    return __builtin_amdgcn_wmma_f32_16x16x4_f32(
        false, a, false, b, (short)0, c, false, false);
}

__device__ __forceinline__ float sigmoidf_(float x) {
    return 1.0f / (1.0f + expf(-x));
}

// ---------------------------------------------------------------------------
// K1: k = h_o_prev @ Wk^T + bk   (N x F), one wave per 16x16 tile, K=256
// ---------------------------------------------------------------------------
__global__ __launch_bounds__(32) void k_gemm_kernel(
    const float* __restrict__ h, const float* __restrict__ Wk,
    const float* __restrict__ bk, float* __restrict__ k_out) {
    const int r0   = blockIdx.x * 16;
    const int c0   = blockIdx.y * 16;
    const int lane = threadIdx.x;
    const int m    = lane & 15;
    const int ksel = (lane >> 4) << 1;   // lanes 0-15 -> K+0/1, lanes 16-31 -> K+2/3

    const float* Arow = h  + (size_t)(r0 + m) * DD;
    const float* Brow = Wk + (size_t)(c0 + m) * DD;   // B[k][n] = Wk[n][k]

    v8f acc = {};
    for (int kk = 0; kk < DD; kk += 4) {
        v2f a; a.x = Arow[kk + ksel]; a.y = Arow[kk + ksel + 1];
        v2f b; b.x = Brow[kk + ksel]; b.y = Brow[kk + ksel + 1];
        acc = wmma4(a, b, acc);
    }
    const int col   = c0 + m;
    const float bia = bk[col];
    const int rbase = r0 + ((lane >> 4) << 3);
    #pragma unroll
    for (int i = 0; i < 8; ++i)
        k_out[(size_t)(rbase + i) * FF + col] = acc[i] + bia;
}

// ---------------------------------------------------------------------------
// K0: beta = softplus(h @ Wb^T + bb) + 1   (N,) -- one wave per row
// ---------------------------------------------------------------------------
__global__ __launch_bounds__(256) void beta_kernel(
    const float* __restrict__ h, const float* __restrict__ Wb,
    const float* __restrict__ bb, float* __restrict__ beta_ws) {
    const int wid  = threadIdx.x >> 5;
    const int lane = threadIdx.x & 31;
    const int row  = blockIdx.x * 8 + wid;
    const float* hr = h + (size_t)row * DD;
    float acc = 0.0f;
    #pragma unroll
    for (int i = 0; i < 8; ++i)
        acc += hr[lane + 32 * i] * Wb[lane + 32 * i];
    #pragma unroll
    for (int m = 16; m >= 1; m >>= 1)
        acc += __shfl_xor(acc, m, 32);
    if (lane == 0) {
        float x  = acc + bb[0];
        float sp = (x > 20.0f) ? x : log1pf(expf(x));
        beta_ws[row] = sp + 1.0f;
    }
}

// ---------------------------------------------------------------------------
// K2: per-row attention. One block per n. C[n] staged once through LDS.
//   s = (C[n] . k[n]) / (||C[n,s]|| * ||k[n]||) * beta -> softmax -> w
//   r = w^T C[n]
// ---------------------------------------------------------------------------
__global__ __launch_bounds__(256) void attn_kernel(
    const float* __restrict__ C, const float* __restrict__ k_in,
    const float* __restrict__ beta_ws, float* __restrict__ w_ws,
    float* __restrict__ r_out) {
    extern __shared__ float ldsC[];          // SS * 129 floats (padded rows)
    __shared__ float ldsK[FF];
    __shared__ float ldsW[SS];
    __shared__ float red[8];
    __shared__ float s_knorm;

    const int n = blockIdx.x;
    const int t = threadIdx.x;

    if (t < FF) ldsK[t] = k_in[(size_t)n * FF + t];

    // Coalesced float4 load of C[n] into padded LDS
    const float4* C4 = (const float4*)(C + (size_t)n * SS * FF);
    #pragma unroll
    for (int i = 0; i < 16; ++i) {
        int idx4 = t + i * 256;              // 4096 float4s total
        float4 cv = C4[idx4];
        int s  = idx4 >> 5;                  // 32 float4 per s-row
        int f0 = (idx4 & 31) << 2;
        float* dst = &ldsC[s * 129 + f0];
        dst[0] = cv.x; dst[1] = cv.y; dst[2] = cv.z; dst[3] = cv.w;
    }
    __syncthreads();

    // ||k||
    float kn = 0.0f;
    if (t < FF) { float kv = ldsK[t]; kn = kv * kv; }
    #pragma unroll
    for (int m = 16; m >= 1; m >>= 1) kn += __shfl_xor(kn, m, 32);
    if ((t & 31) == 0) red[t >> 5] = kn;
    __syncthreads();
    if (t == 0) s_knorm = fmaxf(sqrtf(red[0] + red[1] + red[2] + red[3]), 1e-8f);
    __syncthreads();

    // scores (thread t handles row s=t; stride-129 -> conflict-free banks)
    const float beta = beta_ws[n];
    float sval = -1e30f;
    if (t < SS) {
        const float* row = &ldsC[t * 129];
        float dot = 0.0f, ss = 0.0f;
        for (int f = 0; f < FF; ++f) {
            float cv = row[f];
            dot += cv * ldsK[f];
            ss  += cv * cv;
        }
        float cn = fmaxf(sqrtf(ss), 1e-8f);
        sval = dot / (cn * s_knorm) * beta;
    }

    // softmax across 128 threads (4 waves)
    float mx = sval;
    #pragma unroll
    for (int m = 16; m >= 1; m >>= 1) mx = fmaxf(mx, __shfl_xor(mx, m, 32));
    if ((t & 31) == 0) red[t >> 5] = mx;
    __syncthreads();
    float gmax = fmaxf(fmaxf(red[0], red[1]), fmaxf(red[2], red[3]));
    float ex = (t < SS) ? expf(sval - gmax) : 0.0f;
    float sm = ex;
    #pragma unroll
    for (int m = 16; m >= 1; m >>= 1) sm += __shfl_xor(sm, m, 32);
    __syncthreads();                          // gmax consumed before red reuse
    if ((t & 31) == 0) red[t >> 5] = sm;
    __syncthreads();
    float gsum = red[0] + red[1] + red[2] + red[3];
    if (t < SS) {
        float wv = ex / gsum;
        ldsW[t] = wv;
        w_ws[(size_t)n * SS + t] = wv;
    }
    __syncthreads();

    // r[f] = sum_s w[s] * C[n,s,f]  (thread t handles column f=t)
    if (t < FF) {
        float acc = 0.0f;
        for (int s = 0; s < SS; ++s)
            acc += ldsW[s] * ldsC[s * 129 + t];
        r_out[(size_t)n * FF + t] = acc;
    }
}

// ---------------------------------------------------------------------------
// K3: fused GRU.  gi = r @ Wih^T + bih (K=128), gh = h @ Whh^T + bhh (K=256),
// gates -> h_o.  One wave per (16 rows x 16 cols of D); 6 WMMA accumulators.
// ---------------------------------------------------------------------------
__global__ __launch_bounds__(32) void gru_kernel(
    const float* __restrict__ r_in, const float* __restrict__ h_prev,
    const float* __restrict__ Wih, const float* __restrict__ Whh,
    const float* __restrict__ bih, const float* __restrict__ bhh,
    float* __restrict__ h_o_out) {
    const int r0   = blockIdx.x * 16;
    const int d0   = blockIdx.y * 16;
    const int lane = threadIdx.x;
    const int m    = lane & 15;
    const int ksel = (lane >> 4) << 1;

    const float* Ar = r_in   + (size_t)(r0 + m) * FF;
    const float* Ah = h_prev + (size_t)(r0 + m) * DD;
    const float* B0 = Wih + (size_t)(d0 +   0 + m) * FF;
    const float* B1 = Wih + (size_t)(d0 + 256 + m) * FF;
    const float* B2 = Wih + (size_t)(d0 + 512 + m) * FF;
    const float* H0 = Whh + (size_t)(d0 +   0 + m) * DD;
    const float* H1 = Whh + (size_t)(d0 + 256 + m) * DD;
    const float* H2 = Whh + (size_t)(d0 + 512 + m) * DD;

    v8f gi0 = {}, gi1 = {}, gi2 = {}, gh0 = {}, gh1 = {}, gh2 = {};

    for (int kk = 0; kk < FF; kk += 4) {
        v2f a; a.x = Ar[kk + ksel]; a.y = Ar[kk + ksel + 1];
        v2f b;
        b.x = B0[kk + ksel]; b.y = B0[kk + ksel + 1]; gi0 = wmma4(a, b, gi0);
        b.x = B1[kk + ksel]; b.y = B1[kk + ksel + 1]; gi1 = wmma4(a, b, gi1);
        b.x = B2[kk + ksel]; b.y = B2[kk + ksel + 1]; gi2 = wmma4(a, b, gi2);
    }
    for (int kk = 0; kk < DD; kk += 4) {
        v2f a; a.x = Ah[kk + ksel]; a.y = Ah[kk + ksel + 1];
        v2f b;
        b.x = H0[kk + ksel]; b.y = H0[kk + ksel + 1]; gh0 = wmma4(a, b, gh0);
        b.x = H1[kk + ksel]; b.y = H1[kk + ksel + 1]; gh1 = wmma4(a, b, gh1);
        b.x = H2[kk + ksel]; b.y = H2[kk + ksel + 1]; gh2 = wmma4(a, b, gh2);
    }

    const int d = d0 + m;
    const float bi0 = bih[d], bi1 = bih[d + 256], bi2 = bih[d + 512];
    const float bh0 = bhh[d], bh1 = bhh[d + 256], bh2 = bhh[d + 512];
    const int rbase = r0 + ((lane >> 4) << 3);
    #pragma unroll
    for (int i = 0; i < 8; ++i) {
        float i_r = gi0[i] + bi0, i_z = gi1[i] + bi1, i_n = gi2[i] + bi2;
        float h_r = gh0[i] + bh0, h_z = gh1[i] + bh1, h_n = gh2[i] + bh2;
        float rg = sigmoidf_(i_r + h_r);
        float z  = sigmoidf_(i_z + h_z);
        float nv = tanhf(i_n + rg * h_n);
        float hp = h_prev[(size_t)(rbase + i) * DD + d];
        h_o_out[(size_t)(rbase + i) * DD + d] = (1.0f - z) * nv + z * hp;
    }
}

// ---------------------------------------------------------------------------
// K4: e = sigmoid(h_o @ We^T + be), v = h_o @ Wv^T + bv   (both N x F, K=256)
// ---------------------------------------------------------------------------
__global__ __launch_bounds__(32) void ev_kernel(
    const float* __restrict__ h_o, const float* __restrict__ We,
    const float* __restrict__ be,  const float* __restrict__ Wv,
    const float* __restrict__ bv,  float* __restrict__ e_ws,
    float* __restrict__ v_ws) {
    const int r0   = blockIdx.x * 16;
    const int c0   = blockIdx.y * 16;
    const int lane = threadIdx.x;
    const int m    = lane & 15;
    const int ksel = (lane >> 4) << 1;

    const float* A  = h_o + (size_t)(r0 + m) * DD;
    const float* Be = We  + (size_t)(c0 + m) * DD;
    const float* Bv = Wv  + (size_t)(c0 + m) * DD;

    v8f ae = {}, av = {};
    for (int kk = 0; kk < DD; kk += 4) {
        v2f a; a.x = A[kk + ksel]; a.y = A[kk + ksel + 1];
        v2f b;
        b.x = Be[kk + ksel]; b.y = Be[kk + ksel + 1]; ae = wmma4(a, b, ae);
        b.x = Bv[kk + ksel]; b.y = Bv[kk + ksel + 1]; av = wmma4(a, b, av);
    }
    const int col = c0 + m;
    const float bea = be[col], bva = bv[col];
    const int rbase = r0 + ((lane >> 4) << 3);
    #pragma unroll
    for (int i = 0; i < 8; ++i) {
        size_t o = (size_t)(rbase + i) * FF + col;
        e_ws[o] = sigmoidf_(ae[i] + bea);
        v_ws[o] = av[i] + bva;
    }
}

// ---------------------------------------------------------------------------
// K5: C_new = C * (1 - w*e) + w*v    (pure streaming, float4)
// ---------------------------------------------------------------------------
__global__ __launch_bounds__(256) void cnew_kernel(
    const float* __restrict__ C, const float* __restrict__ w_ws,
    const float* __restrict__ e_ws, const float* __restrict__ v_ws,
    float* __restrict__ C_new) {
    const size_t g = (size_t)blockIdx.x * 256 + threadIdx.x;  // float4 index
    const int n   = (int)(g >> 12);          // 4096 float4 per row n
    const int rem = (int)(g & 4095);
    const int s   = rem >> 5;                // 32 float4 per s-row
    const int fq  = rem & 31;

    const float wv = w_ws[(size_t)n * SS + s];
    const float4 ef = ((const float4*)e_ws)[(size_t)n * 32 + fq];
    const float4 vf = ((const float4*)v_ws)[(size_t)n * 32 + fq];
    const float4 cv = ((const float4*)C)[g];
    float4 o;
    o.x = cv.x * (1.0f - wv * ef.x) + wv * vf.x;
    o.y = cv.y * (1.0f - wv * ef.y) + wv * vf.y;
    o.z = cv.z * (1.0f - wv * ef.z) + wv * vf.z;
    o.w = cv.w * (1.0f - wv * ef.w) + wv * vf.w;
    ((float4*)C_new)[g] = o;
}

// ---------------------------------------------------------------------------
extern "C" void kernel_launch(void* const* d_in, const int* in_sizes, int n_in,
                              void* d_out, int out_size, void* d_ws, size_t ws_size,
                              hipStream_t stream) {
    const float* h_prev = (const float*)d_in[0];
    const float* C      = (const float*)d_in[1];
    const float* Wk     = (const float*)d_in[2];
    const float* bk     = (const float*)d_in[3];
    const float* Wb     = (const float*)d_in[4];
    const float* bb     = (const float*)d_in[5];
    const float* We     = (const float*)d_in[6];
    const float* be     = (const float*)d_in[7];
    const float* Wv     = (const float*)d_in[8];
    const float* bv     = (const float*)d_in[9];
    const float* Wih    = (const float*)d_in[10];
    const float* Whh    = (const float*)d_in[11];
    const float* bih    = (const float*)d_in[12];
    const float* bhh    = (const float*)d_in[13];

    float* out     = (float*)d_out;                 // tuple: h_o, C_new, k, r
    float* h_o_out = out;
    float* C_new   = h_o_out + (size_t)NN * DD;
    float* k_out   = C_new   + (size_t)NN * SS * FF;
    float* r_out   = k_out   + (size_t)NN * FF;

    float* ws      = (float*)d_ws;
    float* w_ws    = ws;                            // N*S
    float* beta_ws = w_ws    + (size_t)NN * SS;     // N
    float* e_ws    = beta_ws + NN;                  // N*F
    float* v_ws    = e_ws    + (size_t)NN * FF;     // N*F

    // K1: k GEMM (WMMA f32)
    k_gemm_kernel<<<dim3(NN / 16, FF / 16), 32, 0, stream>>>(h_prev, Wk, bk, k_out);
    // K0: beta
    beta_kernel<<<NN / 8, 256, 0, stream>>>(h_prev, Wb, bb, beta_ws);
    // K2: attention (one block per row, C staged through LDS once)
    attn_kernel<<<NN, 256, SS * 129 * sizeof(float), stream>>>(C, k_out, beta_ws,
                                                               w_ws, r_out);
    // K3: fused GRU (WMMA f32)
    gru_kernel<<<dim3(NN / 16, DD / 16), 32, 0, stream>>>(r_out, h_prev, Wih, Whh,
                                                          bih, bhh, h_o_out);
    // K4: e / v GEMMs (WMMA f32)
    ev_kernel<<<dim3(NN / 16, FF / 16), 32, 0, stream>>>(h_o_out, We, be, Wv, bv,
                                                         e_ws, v_ws);
    // K5: C_new streaming update
    cnew_kernel<<<(NN * SS * FF / 4) / 256, 256, 0, stream>>>(C, w_ws, e_ws, v_ws,
                                                              C_new);
}